// GCN_28595892257551
// MI455X (gfx1250) — compile-verified
//
#include <hip/hip_runtime.h>

typedef __attribute__((ext_vector_type(2))) float v2f;
typedef __attribute__((ext_vector_type(8))) float v8f;

// ---------------- degree / normalization ----------------

__global__ void k_deg_init(float* __restrict__ deg, int n) {
    int i = blockIdx.x * blockDim.x + threadIdx.x;
    if (i < n) deg[i] = 1.0f;   // self-loop
}

__global__ void k_deg_count(float* __restrict__ deg, const int* __restrict__ dst, int e) {
    int i = blockIdx.x * blockDim.x + threadIdx.x;
    if (i < e) unsafeAtomicAdd(&deg[dst[i]], 1.0f);
}

__global__ void k_dinv(float* __restrict__ deg, int n) {
    int i = blockIdx.x * blockDim.x + threadIdx.x;
    if (i < n) deg[i] = 1.0f / sqrtf(deg[i]);   // deg >= 1 always
}

// ---------------- fp32 WMMA GEMM: C[N x NC] = A[N x K] * B[K x NC] ----------------
// One wave computes one 16x16 tile of C. blockDim.x = 32 * (NC/16).
// blockIdx.x selects the 16-row band; wave id selects the 16-col band.
// Requires N % 16 == 0 (true: 100000 = 6250*16), so EXEC is all-ones at WMMA.

__global__ __launch_bounds__(256) void k_gemm_wmma_f32(
    const float* __restrict__ A, const float* __restrict__ B, float* __restrict__ C,
    int K, int NC) {
    const int lane = threadIdx.x & 31;
    const int wave = threadIdx.x >> 5;
    const int row0 = blockIdx.x * 16;
    const int col0 = wave * 16;
    const int mn   = lane & 15;          // M for A-frag, N for B/D-frags
    const int khalf = (lane >> 4) * 2;   // K sub-offset: 0 (lanes 0-15) or 2 (lanes 16-31)

    const float* __restrict__ arow = A + (size_t)(row0 + mn) * K;
    v8f acc = {};

    for (int k = 0; k < K; k += 4) {
        // A fragment (16x4): lane holds A[m][k+khalf .. k+khalf+1]
        v2f a;
        a.x = arow[k + khalf];
        a.y = arow[k + khalf + 1];
        // B fragment (4x16): lane holds B[k+khalf][n], B[k+khalf+1][n]
        v2f b;
        b.x = B[(size_t)(k + khalf)     * NC + col0 + mn];
        b.y = B[(size_t)(k + khalf + 1) * NC + col0 + mn];
        acc = __builtin_amdgcn_wmma_f32_16x16x4_f32(
            /*neg_a=*/false, a, /*neg_b=*/false, b,
            /*c_mod=*/(short)0, acc, /*reuse_a=*/false, /*reuse_b=*/false);
    }

    // D layout: VGPR r -> M = r + 8*(lane>>4), N = lane&15
    const int mbase = row0 + (lane >> 4) * 8;
#pragma unroll
    for (int r = 0; r < 8; ++r)
        C[(size_t)(mbase + r) * NC + col0 + mn] = acc[r];
}

// ---------------- aggregation ----------------

// out[i,f] = h[i,f] * dinv[i]^2 + bias[f]   (self-loop term + bias)
__global__ void k_agg_init(float* __restrict__ out, const float* __restrict__ h,
                           const float* __restrict__ dinv, const float* __restrict__ bias,
                           int total, int NF) {
    int idx = blockIdx.x * blockDim.x + threadIdx.x;
    if (idx >= total) return;
    int i = idx / NF, f = idx - i * NF;
    float di = dinv[i];
    out[idx] = h[idx] * di * di + bias[f];
}

// One wave per edge; each lane handles VEC contiguous features.
// VEC = NF/32 (4 for NF=128, 2 for NF=64).
template <int VEC>
__global__ __launch_bounds__(256) void k_agg_edges(
    float* __restrict__ out, const float* __restrict__ h,
    const int* __restrict__ src, const int* __restrict__ dst,
    const float* __restrict__ dinv, int E, int NF) {
    int e = blockIdx.x * (blockDim.x >> 5) + (threadIdx.x >> 5);
    if (e >= E) return;
    const int lane = threadIdx.x & 31;
    const int s = src[e];
    const int d = dst[e];
    const float norm = dinv[s] * dinv[d];
    const float* __restrict__ hp = h + (size_t)s * NF + lane * VEC;
    float* __restrict__ op = out + (size_t)d * NF + lane * VEC;
#pragma unroll
    for (int i = 0; i < VEC; ++i)
        unsafeAtomicAdd(&op[i], hp[i] * norm);
}

__global__ void k_relu(float* __restrict__ x, int n) {
    int i = blockIdx.x * blockDim.x + threadIdx.x;
    if (i < n) x[i] = fmaxf(x[i], 0.0f);
}

// ---------------- launch ----------------

extern "C" void kernel_launch(void* const* d_in, const int* in_sizes, int n_in,
                              void* d_out, int out_size, void* d_ws, size_t ws_size,
                              hipStream_t stream) {
    const int D_IN = 512, D_LAT = 128, D_OUT = 64;

    const float* x  = (const float*)d_in[0];
    const int*   ei = (const int*)d_in[1];
    const float* W1 = (const float*)d_in[2];
    const float* b1 = (const float*)d_in[3];
    const float* W2 = (const float*)d_in[4];
    const float* b2 = (const float*)d_in[5];
    float* out = (float*)d_out;

    const int N = in_sizes[0] / D_IN;   // 100000
    const int E = in_sizes[1] / 2;      // 1600000
    const int* src = ei;
    const int* dst = ei + E;

    // workspace carve-up (all fp32), 256B-aligned segments
    char* ws = (char*)d_ws;
    size_t off = 0;
    auto carve = [&](size_t nfloats) {
        float* p = (float*)(ws + off);
        off += ((nfloats * sizeof(float)) + 255) & ~(size_t)255;
        return p;
    };
    float* dinv = carve((size_t)N);            // degrees -> dinv (in place)
    float* h1   = carve((size_t)N * D_LAT);    // x @ W1
    float* a1   = carve((size_t)N * D_LAT);    // layer-1 aggregated output
    float* h2   = carve((size_t)N * D_OUT);    // a1 @ W2

    const int T = 256;

    // 1) symmetric normalization coefficients
    k_deg_init<<<(N + T - 1) / T, T, 0, stream>>>(dinv, N);
    k_deg_count<<<(E + T - 1) / T, T, 0, stream>>>(dinv, dst, E);
    k_dinv<<<(N + T - 1) / T, T, 0, stream>>>(dinv, N);

    // 2) layer 1: h1 = x @ W1  (K=512, NC=128 -> 8 waves/block)
    k_gemm_wmma_f32<<<N / 16, 32 * (D_LAT / 16), 0, stream>>>(x, W1, h1, D_IN, D_LAT);

    // 3) layer 1 aggregation: a1 = scatter(norm * h1[src]) + selfloop + b1
    k_agg_init<<<((size_t)N * D_LAT + T - 1) / T, T, 0, stream>>>(a1, h1, dinv, b1, N * D_LAT, D_LAT);
    k_agg_edges<4><<<(E + 7) / 8, T, 0, stream>>>(a1, h1, src, dst, dinv, E, D_LAT);

    // 4) layer 2: h2 = a1 @ W2  (K=128, NC=64 -> 4 waves/block)
    k_gemm_wmma_f32<<<N / 16, 32 * (D_OUT / 16), 0, stream>>>(a1, W2, h2, D_LAT, D_OUT);

    // 5) layer 2 aggregation directly into d_out, then ReLU
    k_agg_init<<<((size_t)N * D_OUT + T - 1) / T, T, 0, stream>>>(out, h2, dinv, b2, N * D_OUT, D_OUT);
    k_agg_edges<2><<<(E + 7) / 8, T, 0, stream>>>(out, h2, src, dst, dinv, E, D_OUT);
    k_relu<<<(N * D_OUT + T - 1) / T, T, 0, stream>>>(out, N * D_OUT);
}